// FCN_ManualFeats_Early_Fusion_37099927503292
// MI455X (gfx1250) — compile-verified
//
#include <hip/hip_runtime.h>
#include <hip/hip_bf16.h>

// ---------------------------------------------------------------------------
// 15-block MLP: Linear -> ReLU -> BatchNorm1d (batch stats) per block.
// Per block: kernel A (WMMA f32 GEMM + bias + relu + partial column stats),
// kernel R (reduce stats -> scale/shift), kernel B (elementwise normalize).
// fp32 WMMA (v_wmma_f32_16x16x4_f32) keeps reference precision; workload is
// L2-resident (each tensor ~32MB << 192MB L2), so memory path dominates.
// All reductions are fixed-order (no float atomics) => deterministic output.
// ---------------------------------------------------------------------------

typedef __attribute__((ext_vector_type(2))) float v2f;
typedef __attribute__((ext_vector_type(8))) float v8f;

#define TILE_M     128
#define NW         8          // waves per block (256 threads, wave32)
#define LDSS       68         // LDS row stride in floats (68 % 64 == 4 -> conflict-free)

// ---------------- Kernel A: GEMM + bias + ReLU + per-block column stats ----
__global__ __launch_bounds__(256)
void gemm_relu_stats(const float* __restrict__ xin, int ldx, int Kp,
                     const float* __restrict__ W,  int Kw,
                     const float* __restrict__ bias,
                     float* __restrict__ hout,          // [B,64] padded
                     float* __restrict__ partial)       // [nblk,128] sums|sumsq
{
    __shared__ float xl[TILE_M * LDSS];
    __shared__ float Wl[64 * LDSS];
    __shared__ float biasl[64];
    __shared__ float wsum[NW * 64];
    __shared__ float wsq [NW * 64];

    const int t    = threadIdx.x;
    const int lane = t & 31;
    const int wave = t >> 5;
    const int half = lane >> 4;
    const int l15  = lane & 15;
    const size_t row0 = (size_t)blockIdx.x * TILE_M;

    // ---- stage x tile (coalesced b128 loads) ----
    const int c4n = Kp >> 2;                       // float4s per row
    for (int i = t; i < TILE_M * c4n; i += 256) {
        int r = i / c4n;
        int c = (i - r * c4n) * 4;
        float4 v = *(const float4*)(xin + (row0 + r) * (size_t)ldx + c);
        *(float4*)(&xl[r * LDSS + c]) = v;
    }
    // ---- stage W zero-padded to 64 x Kp ----
    for (int i = t; i < 64 * Kp; i += 256) {
        int n = i / Kp;
        int k = i - n * Kp;
        Wl[n * LDSS + k] = (n < 59 && k < Kw) ? W[n * Kw + k] : 0.0f;
    }
    if (t < 64) biasl[t] = (t < 59) ? bias[t] : 0.0f;
    __syncthreads();

    // ---- WMMA main loop: this wave computes rows [wave*16, wave*16+16) x 64 cols
    v8f acc[4] = {};
    const float* ax = xl + (wave * 16 + l15) * LDSS;
    for (int k0 = 0; k0 < Kp; k0 += 4) {
        const int ka = k0 + 2 * half;              // A/B frag: k = 2*half + vgpr
        v2f a = *(const v2f*)(ax + ka);
        #pragma unroll
        for (int tt = 0; tt < 4; tt++) {
            v2f b = *(const v2f*)(Wl + (tt * 16 + l15) * LDSS + ka);
            acc[tt] = __builtin_amdgcn_wmma_f32_16x16x4_f32(
                false, a, false, b, (short)0, acc[tt], false, false);
        }
    }

    // ---- bias + ReLU + store + per-wave column sums ----
    const size_t mbase = row0 + wave * 16 + half * 8;   // M = r + 8*half
    #pragma unroll
    for (int tt = 0; tt < 4; tt++) {
        const int n = tt * 16 + l15;
        const float bn = biasl[n];
        float sl = 0.f, sq = 0.f;
        #pragma unroll
        for (int r = 0; r < 8; r++) {
            float v = acc[tt][r] + bn;
            v = fmaxf(v, 0.f);
            hout[(mbase + r) * 64 + n] = v;
            sl += v;
            sq += v * v;
        }
        sl += __shfl_xor(sl, 16, 32);   // combine M-halves (lane ^ 16)
        sq += __shfl_xor(sq, 16, 32);
        if (half == 0) {                // lanes 0..15 now hold 16-row col sums
            wsum[wave * 64 + n] = sl;
            wsq [wave * 64 + n] = sq;
        }
    }
    __syncthreads();

    // ---- fixed-order cross-wave reduction -> per-block partials ----
    if (t < 64) {
        float s = 0.f;
        #pragma unroll
        for (int w = 0; w < NW; w++) s += wsum[w * 64 + t];
        partial[(size_t)blockIdx.x * 128 + t] = s;
    } else if (t < 128) {
        const int n = t - 64;
        float s = 0.f;
        #pragma unroll
        for (int w = 0; w < NW; w++) s += wsq[w * 64 + n];
        partial[(size_t)blockIdx.x * 128 + 64 + n] = s;
    }
}

// ---------------- Kernel R: reduce partials -> scale/shift -----------------
__global__ void stats_reduce(const float* __restrict__ partial, int nblk,
                             const float* __restrict__ g,
                             const float* __restrict__ bt,
                             float* __restrict__ stats,   // scale[64], shift[64]
                             float invB)
{
    const int n = threadIdx.x;
    if (n >= 64) return;
    float s = 0.f, ss = 0.f;
    for (int b = 0; b < nblk; b++) {                 // fixed order: deterministic
        s  += partial[(size_t)b * 128 + n];
        ss += partial[(size_t)b * 128 + 64 + n];
    }
    const float mu  = s * invB;
    float var = ss * invB - mu * mu;
    var = fmaxf(var, 0.f);
    float scale = 0.f, shift = 0.f;
    if (n < 59) {
        scale = g[n] * rsqrtf(var + 1e-5f);
        shift = bt[n] - mu * scale;
    }
    stats[n]      = scale;   // padded cols get scale=shift=0 -> stay zero
    stats[64 + n] = shift;
}

// ---------------- Kernel B: normalize (padded layout, feeds next GEMM) -----
__global__ __launch_bounds__(256)
void normalize_pad(const float* __restrict__ h, const float* __restrict__ stats,
                   float* __restrict__ out, size_t nvec /* = B*16 float4s */)
{
    __shared__ float sc[64], sh[64];
    const int t = threadIdx.x;
    if (t < 64)       sc[t]      = stats[t];
    else if (t < 128) sh[t - 64] = stats[t];
    __syncthreads();

    const size_t idx = (size_t)blockIdx.x * 256 + t;
    if (idx >= nvec) return;
    const int n = (int)(idx & 15) * 4;
    const float4 hv = ((const float4*)h)[idx];
    float4 o;
    o.x = hv.x * sc[n + 0] + sh[n + 0];
    o.y = hv.y * sc[n + 1] + sh[n + 1];
    o.z = hv.z * sc[n + 2] + sh[n + 2];
    o.w = hv.w * sc[n + 3] + sh[n + 3];
    ((float4*)out)[idx] = o;
}

// ---------------- Final normalize: strip padding, write [B,59] -------------
__global__ void normalize_final(const float* __restrict__ h,
                                const float* __restrict__ stats,
                                float* __restrict__ out, size_t total /* B*59 */)
{
    const size_t i = (size_t)blockIdx.x * 256 + threadIdx.x;
    if (i >= total) return;
    const size_t row = i / 59;
    const int n = (int)(i - row * 59);
    out[i] = h[row * 64 + n] * stats[n] + stats[64 + n];
}

// ---------------- Pad input x [B,29] -> xpad [B,32] ------------------------
__global__ void pad_input(const float* __restrict__ x, float* __restrict__ xp,
                          size_t total /* B*32 */)
{
    const size_t i = (size_t)blockIdx.x * 256 + threadIdx.x;
    if (i >= total) return;
    const size_t row = i >> 5;
    const int n = (int)(i & 31);
    xp[i] = (n < 29) ? x[row * 29 + n] : 0.0f;
}

// ---------------------------------------------------------------------------
extern "C" void kernel_launch(void* const* d_in, const int* in_sizes, int n_in,
                              void* d_out, int out_size, void* d_ws, size_t ws_size,
                              hipStream_t stream) {
    const float* x   = (const float*)d_in[0];   // [B,29]
    const float* W0  = (const float*)d_in[1];   // [59,29]
    const float* b0  = (const float*)d_in[2];   // [59]
    const float* g0  = (const float*)d_in[3];   // [59]
    const float* bt0 = (const float*)d_in[4];   // [59]
    const float* Ws  = (const float*)d_in[5];   // [14,59,59]
    const float* bs  = (const float*)d_in[6];   // [14,59]
    const float* gs  = (const float*)d_in[7];   // [14,59]
    const float* bts = (const float*)d_in[8];   // [14,59]
    float* out = (float*)d_out;

    const int B     = in_sizes[0] / 29;         // 131072
    const int nblk  = B / TILE_M;               // GEMM workgroups (1024)
    const float invB = 1.0f / (float)B;

    float* ws      = (float*)d_ws;
    float* xpad    = ws;                                  // B*32
    float* P       = xpad    + (size_t)B * 32;            // B*64 (h buffer)
    float* Q       = P       + (size_t)B * 64;            // B*64 (y buffer)
    float* partial = Q       + (size_t)B * 64;            // nblk*128
    float* stats   = partial + (size_t)nblk * 128;        // 128

    const size_t padTot = (size_t)B * 32;
    pad_input<<<(unsigned)((padTot + 255) / 256), 256, 0, stream>>>(x, xpad, padTot);

    const size_t nvec = (size_t)B * 16;                   // float4s in [B,64]

    // ---- block 0: Linear(29,59) + ReLU + BN ----
    gemm_relu_stats<<<nblk, 256, 0, stream>>>(xpad, 32, 32, W0, 29, b0, P, partial);
    stats_reduce<<<1, 64, 0, stream>>>(partial, nblk, g0, bt0, stats, invB);
    normalize_pad<<<(unsigned)(nvec / 256), 256, 0, stream>>>(P, stats, Q, nvec);

    // ---- blocks 1..14: Linear(59,59) + ReLU + BN ----
    for (int l = 0; l < 14; l++) {
        const float* W  = Ws  + (size_t)l * 59 * 59;
        const float* bb = bs  + (size_t)l * 59;
        const float* gg = gs  + (size_t)l * 59;
        const float* tt = bts + (size_t)l * 59;
        gemm_relu_stats<<<nblk, 256, 0, stream>>>(Q, 64, 64, W, 59, bb, P, partial);
        stats_reduce<<<1, 64, 0, stream>>>(partial, nblk, gg, tt, stats, invB);
        if (l < 13) {
            normalize_pad<<<(unsigned)(nvec / 256), 256, 0, stream>>>(P, stats, Q, nvec);
        } else {
            const size_t total = (size_t)B * 59;
            normalize_final<<<(unsigned)((total + 255) / 256), 256, 0, stream>>>(
                P, stats, out, total);
        }
    }
}